// GaugeLatticeLayer_76364518523596
// MI455X (gfx1250) — compile-verified
//
#include <hip/hip_runtime.h>

#define N_NODES 65536
#define N_EDGES 1048576
#define MUL_S 32
#define MUL_V 16
#define X_DIM 80          // MUL_S + 3*MUL_V
#define EPS_BN 1e-5f

typedef __attribute__((ext_vector_type(2))) float v2f;
typedef __attribute__((ext_vector_type(8))) float v8f;

// ---------------------------------------------------------------------------
// V_WMMA_F32_16X16X4_F32 wrapper.
// A: 16x4 f32 (2 VGPR/lane): lane L -> row m=L&15, half h=L>>4, a={A[m][k0+2h],A[m][k0+2h+1]}
// B: 4x16  f32 (2 VGPR/lane): lane L -> col n=L&15, b={B[k0+2h][n],B[k0+2h+1][n]}
// C/D: 16x16 f32 (8 VGPR): vgpr r -> row m=r+8h, col n=L&15
// ---------------------------------------------------------------------------
__device__ __forceinline__ v8f wmma_k4(v2f a, v2f b, v8f c) {
  return __builtin_amdgcn_wmma_f32_16x16x4_f32(
      /*neg_a=*/false, a, /*neg_b=*/false, b,
      /*c_mod=*/(short)0, c, /*reuse_a=*/false, /*reuse_b=*/false);
}

// ---------------------------------------------------------------------------
// Kernel 0: zero the stats scratch (160 floats in d_ws)
// ---------------------------------------------------------------------------
__global__ void zero_stats_kernel(float* __restrict__ stats) {
  if (threadIdx.x < 160) stats[threadIdx.x] = 0.0f;
}

// ---------------------------------------------------------------------------
// Kernel 1: self-interaction.  One wave handles 16 nodes.
//   ts = xs @ W_self_s / sqrt(32)   (K=32, N=32 -> 2 tiles)
//   tv_c = xv_c @ W_self_v / sqrt(16)  for c=0..2
// Weights staged in LDS as pre-paired v2f B operands (one ds_load_b64 each).
// Plain stores initialize the full output/accumulator array.
// ---------------------------------------------------------------------------
__global__ __launch_bounds__(256) void self_kernel(
    const float* __restrict__ x, const float* __restrict__ Ws,
    const float* __restrict__ Wv, float* __restrict__ out) {
  __shared__ v2f lWs[16 * 32];  // [kp][n], kp = k/2 (K=32), n = 0..31
  __shared__ v2f lWv[8 * 16];   // [kp][n], kp = k/2 (K=16), n = 0..15

  for (int i = threadIdx.x; i < 16 * 32; i += 256) {
    const int kp = i >> 5, nn = i & 31;
    v2f w; w.x = Ws[(2 * kp) * MUL_S + nn]; w.y = Ws[(2 * kp + 1) * MUL_S + nn];
    lWs[i] = w;
  }
  for (int i = threadIdx.x; i < 8 * 16; i += 256) {
    const int kp = i >> 4, nn = i & 15;
    v2f w; w.x = Wv[(2 * kp) * MUL_V + nn]; w.y = Wv[(2 * kp + 1) * MUL_V + nn];
    lWv[i] = w;
  }
  __syncthreads();

  const int lane = threadIdx.x & 31;
  const int wave = blockIdx.x * 8 + (threadIdx.x >> 5);
  const int node0 = wave * 16;
  const int m = lane & 15;
  const int h = lane >> 4;
  const int n = lane & 15;
  const float* __restrict__ xrow = x + (size_t)(node0 + m) * X_DIM;

  // ts tile: 16 nodes x 32 channels
  v8f acc0 = {}; v8f acc1 = {};
#pragma unroll
  for (int k0 = 0; k0 < MUL_S; k0 += 4) {
    const int k = k0 + 2 * h;
    v2f a; a.x = xrow[k]; a.y = xrow[k + 1];
    acc0 = wmma_k4(a, lWs[(k >> 1) * 32 + n], acc0);
    acc1 = wmma_k4(a, lWs[(k >> 1) * 32 + 16 + n], acc1);
  }
  const float ss = 0.17677669529663687f;  // 1/sqrt(32)
#pragma unroll
  for (int r = 0; r < 8; ++r) {
    float* __restrict__ orow = out + (size_t)(node0 + r + 8 * h) * X_DIM;
    orow[n]      = acc0[r] * ss;
    orow[16 + n] = acc1[r] * ss;
  }

  // tv tiles: 3 components, 16 nodes x 16 channels each
  const float sv = 0.25f;  // 1/sqrt(16)
#pragma unroll
  for (int c = 0; c < 3; ++c) {
    v8f acc = {};
#pragma unroll
    for (int k0 = 0; k0 < MUL_V; k0 += 4) {
      const int k = k0 + 2 * h;
      v2f a; a.x = xrow[MUL_S + k * 3 + c]; a.y = xrow[MUL_S + (k + 1) * 3 + c];
      acc = wmma_k4(a, lWv[(k >> 1) * 16 + n], acc);
    }
#pragma unroll
    for (int r = 0; r < 8; ++r) {
      float* __restrict__ orow = out + (size_t)(node0 + r + 8 * h) * X_DIM;
      orow[MUL_S + n * 3 + c] = acc[r] * sv;
    }
  }
}

// ---------------------------------------------------------------------------
// Kernel 2: edge messages.  One wave handles 16 edges (M-tile).  Weights are
// staged once per block in LDS as paired v2f B operands; the dots-GEMM and
// cross-GEMM share a single pass over xv_src (5 WMMAs / K-step, 5 independent
// accumulators).  Results scatter-added into out[dst] with f32 atomics.
// ---------------------------------------------------------------------------
__global__ __launch_bounds__(256) void edge_kernel(
    const float* __restrict__ x, const int* __restrict__ ei32,
    const float* __restrict__ eattr,
    const float* __restrict__ Wsv,   // 32x16
    const float* __restrict__ Wvs,   // 16x32
    const float* __restrict__ Wvv,   // 16x16
    float* __restrict__ out) {
  __shared__ v2f  lWsv[16 * 16];  // [kp][n] kp<16 (K=32, N=16)
  __shared__ v2f  lWvs[8 * 32];   // [kp][n] kp<8  (K=16, N=32)
  __shared__ v2f  lWvv[8 * 16];   // [kp][n] kp<8  (K=16, N=16)
  __shared__ float s_attr[8][16][3];
  __shared__ int   s_dst[8][16];

  for (int i = threadIdx.x; i < 16 * 16; i += 256) {
    const int kp = i >> 4, nn = i & 15;
    v2f w; w.x = Wsv[(2 * kp) * MUL_V + nn]; w.y = Wsv[(2 * kp + 1) * MUL_V + nn];
    lWsv[i] = w;
  }
  for (int i = threadIdx.x; i < 8 * 32; i += 256) {
    const int kp = i >> 5, nn = i & 31;
    v2f w; w.x = Wvs[(2 * kp) * MUL_S + nn]; w.y = Wvs[(2 * kp + 1) * MUL_S + nn];
    lWvs[i] = w;
  }
  for (int i = threadIdx.x; i < 8 * 16; i += 256) {
    const int kp = i >> 4, nn = i & 15;
    v2f w; w.x = Wvv[(2 * kp) * MUL_V + nn]; w.y = Wvv[(2 * kp + 1) * MUL_V + nn];
    lWvv[i] = w;
  }

  const int lane = threadIdx.x & 31;
  const int wid  = threadIdx.x >> 5;
  const int e0   = (blockIdx.x * 8 + wid) * 16;
  const int m = lane & 15;
  const int h = lane >> 4;
  const int n = lane & 15;
  const int e = e0 + m;

  const int src = ei32[2 * e];  // low dword of int64 index
  if (h == 0) {
    s_dst[wid][m]     = ei32[2 * (N_EDGES + e)];
    s_attr[wid][m][0] = eattr[(size_t)e * 3 + 0];
    s_attr[wid][m][1] = eattr[(size_t)e * 3 + 1];
    s_attr[wid][m][2] = eattr[(size_t)e * 3 + 2];
  }
  __syncthreads();

  const float ea0 = s_attr[wid][m][0];
  const float ea1 = s_attr[wid][m][1];
  const float ea2 = s_attr[wid][m][2];
  const float* __restrict__ xsrc = x + (size_t)src * X_DIM;

  // ---- G1 = xs_src @ W_tp_sv     (16x32 @ 32x16 -> 16x16) ----
  v8f g1 = {};
#pragma unroll
  for (int k0 = 0; k0 < MUL_S; k0 += 4) {
    const int k = k0 + 2 * h;
    v2f a; a.x = xsrc[k]; a.y = xsrc[k + 1];
    g1 = wmma_k4(a, lWsv[(k >> 1) * 16 + n], g1);
  }

  // ---- merged: dots @ W_tp_vs  and  cross_c @ W_tp_vv (single xv pass) ----
  v8f ms0 = {}, ms1 = {};
  v8f gc0 = {}, gc1 = {}, gc2 = {};
#pragma unroll
  for (int k0 = 0; k0 < MUL_V; k0 += 4) {
    const int k = k0 + 2 * h;
    const int kp = k >> 1;
    const float* __restrict__ p = xsrc + MUL_S + k * 3;
    const float u0x = p[0], u0y = p[1], u0z = p[2];
    const float u1x = p[3], u1y = p[4], u1z = p[5];
    v2f ad;                                   // dots
    ad.x = u0x * ea0 + u0y * ea1 + u0z * ea2;
    ad.y = u1x * ea0 + u1y * ea1 + u1z * ea2;
    v2f a0; a0.x = u0y * ea2 - u0z * ea1; a0.y = u1y * ea2 - u1z * ea1;  // cross x
    v2f a1; a1.x = u0z * ea0 - u0x * ea2; a1.y = u1z * ea0 - u1x * ea2;  // cross y
    v2f a2; a2.x = u0x * ea1 - u0y * ea0; a2.y = u1x * ea1 - u1y * ea0;  // cross z
    ms0 = wmma_k4(ad, lWvs[kp * 32 + n], ms0);
    ms1 = wmma_k4(ad, lWvs[kp * 32 + 16 + n], ms1);
    v2f bv = lWvv[kp * 16 + n];
    gc0 = wmma_k4(a0, bv, gc0);
    gc1 = wmma_k4(a1, bv, gc1);
    gc2 = wmma_k4(a2, bv, gc2);
  }

  // ---- scatter with all scale factors folded in ----
  const float s_ms = 0.14433756729740646f;  // 1/(sqrt(3)*sqrt(16))
  const float s_mv = 0.14433756729740646f;  // 1/sqrt(48)
  const float s_cr = 0.70710678118654752f;  // 1/sqrt(2)
#pragma unroll
  for (int r = 0; r < 8; ++r) {
    const int mm = r + 8 * h;
    const int d  = s_dst[wid][mm];
    const float a0 = s_attr[wid][mm][0];
    const float a1 = s_attr[wid][mm][1];
    const float a2 = s_attr[wid][mm][2];
    float* __restrict__ orow = out + (size_t)d * X_DIM;
    atomicAdd(&orow[n],      ms0[r] * s_ms);
    atomicAdd(&orow[16 + n], ms1[r] * s_ms);
    const float g = g1[r];
    atomicAdd(&orow[MUL_S + n * 3 + 0], (g * a0 + s_cr * gc0[r]) * s_mv);
    atomicAdd(&orow[MUL_S + n * 3 + 1], (g * a1 + s_cr * gc1[r]) * s_mv);
    atomicAdd(&orow[MUL_S + n * 3 + 2], (g * a2 + s_cr * gc2[r]) * s_mv);
  }
}

// ---------------------------------------------------------------------------
// Kernel 3: per-channel sums for batchnorm.  Each block reduces 256 nodes;
// thread t (t < 80) owns channel t.  Partial sums atomically added to stats.
// stats layout (floats): [0,32) sum_s | [32,64) sumsq_s | [64,80) sumsq_v
//                        [80,112) scale_s | [112,144) off_s | [144,160) scale_v
// ---------------------------------------------------------------------------
__global__ __launch_bounds__(128) void stats_kernel(
    const float* __restrict__ out, float* __restrict__ stats) {
  const int ch = threadIdx.x;
  if (ch >= X_DIM) return;
  const int node0 = blockIdx.x * 256;
  float s = 0.0f, sq = 0.0f;
  for (int i = 0; i < 256; ++i) {
    const float v = out[(size_t)(node0 + i) * X_DIM + ch];
    s  += v;
    sq += v * v;
  }
  if (ch < MUL_S) {
    atomicAdd(&stats[ch], s);
    atomicAdd(&stats[32 + ch], sq);
  } else {
    atomicAdd(&stats[64 + (ch - MUL_S) / 3], sq);
  }
}

// ---------------------------------------------------------------------------
// Kernel 4: turn raw sums into per-channel scale/offset.
// ---------------------------------------------------------------------------
__global__ void finalize_scales_kernel(float* __restrict__ stats,
                                       const float* __restrict__ bnw,
                                       const float* __restrict__ bnb) {
  const int t = threadIdx.x;
  const float invn = 1.0f / (float)N_NODES;
  if (t < MUL_S) {
    const float mu  = stats[t] * invn;
    const float var = stats[32 + t] * invn - mu * mu;
    const float sc  = rsqrtf(var + EPS_BN) * bnw[t];
    stats[80 + t]  = sc;
    stats[112 + t] = bnb[t] - mu * sc;
  } else if (t < MUL_S + MUL_V) {
    const int u = t - MUL_S;
    const float nv = stats[64 + u] * (invn / 3.0f);
    stats[144 + u] = rsqrtf(nv + EPS_BN) * bnw[MUL_S + u];
  }
}

// ---------------------------------------------------------------------------
// Kernel 5: apply batchnorm in place.
// ---------------------------------------------------------------------------
__global__ __launch_bounds__(256) void apply_kernel(
    float* __restrict__ out, const float* __restrict__ stats) {
  const size_t tid = (size_t)blockIdx.x * blockDim.x + threadIdx.x;
  if (tid >= (size_t)N_NODES * X_DIM) return;
  const int ch = (int)(tid % X_DIM);
  const float v = out[tid];
  if (ch < MUL_S) {
    out[tid] = v * stats[80 + ch] + stats[112 + ch];
  } else {
    out[tid] = v * stats[144 + (ch - MUL_S) / 3];
  }
}

// ---------------------------------------------------------------------------
extern "C" void kernel_launch(void* const* d_in, const int* in_sizes, int n_in,
                              void* d_out, int out_size, void* d_ws, size_t ws_size,
                              hipStream_t stream) {
  const float* x     = (const float*)d_in[0];
  const int*   ei32  = (const int*)d_in[1];   // int64 indices, read low dwords
  const float* eattr = (const float*)d_in[2];
  const float* Ws    = (const float*)d_in[3];
  const float* Wv    = (const float*)d_in[4];
  const float* Wsv   = (const float*)d_in[5];
  const float* Wvs   = (const float*)d_in[6];
  const float* Wvv   = (const float*)d_in[7];
  const float* bnw   = (const float*)d_in[8];
  const float* bnb   = (const float*)d_in[9];
  float* out   = (float*)d_out;
  float* stats = (float*)d_ws;

  zero_stats_kernel<<<1, 256, 0, stream>>>(stats);
  // 65536 nodes / 16 per wave = 4096 waves / 8 waves per block = 512 blocks
  self_kernel<<<512, 256, 0, stream>>>(x, Ws, Wv, out);
  // 1048576 edges / 16 per wave = 65536 waves / 8 = 8192 blocks
  edge_kernel<<<8192, 256, 0, stream>>>(x, ei32, eattr, Wsv, Wvs, Wvv, out);
  // 65536 nodes / 256 per block = 256 blocks
  stats_kernel<<<256, 128, 0, stream>>>(out, stats);
  finalize_scales_kernel<<<1, 64, 0, stream>>>(stats, bnw, bnb);
  // 65536*80 elements / 256 = 20480 blocks
  apply_kernel<<<20480, 256, 0, stream>>>(out, stats);
}